// Task_Recog_Model_8392366096779
// MI455X (gfx1250) — compile-verified
//
#include <hip/hip_runtime.h>
#include <hip/hip_bf16.h>

typedef float v2f __attribute__((ext_vector_type(2)));
typedef float v8f __attribute__((ext_vector_type(8)));

#define NFEAT 124
#define HID 3
#define OUT 6
#define BSZ 1024
#define TLEN 512
#define MROWS (BSZ * TLEN)   // 524288 flattened (b,t) rows

// ---------------------------------------------------------------------------
// Phase 1: z0[m][h] = x[m,:] @ W_ih0[h,:] + b_ih0[h] + b_hh0[h]
// Skinny GEMM M=524288, K=124, N=3 via V_WMMA_F32_16X16X4_F32.
// One 16-row tile per wave, 31 K-chunks of 4. Bias folded into C init.
// ---------------------------------------------------------------------------
__global__ void __launch_bounds__(256)
rnn_phase1_wmma(const float* __restrict__ x,
                const float* __restrict__ Wih0,
                const float* __restrict__ bih0,
                const float* __restrict__ bhh0,
                float* __restrict__ z0) {
    const int lane = threadIdx.x & 31;
    const int wave = threadIdx.x >> 5;
    const int tile = (blockIdx.x * (blockDim.x >> 5) + wave) * 16;  // first M row
    const int half = lane >> 4;    // 0: K pair {0,1}, 1: K pair {2,3}
    const int l16  = lane & 15;    // A: M row within tile; B/C: N column

    // C init = bias for column n = l16 (only n<3 is real)
    const bool ncol = (l16 < HID);
    const float bias = ncol ? (bih0[l16] + bhh0[l16]) : 0.0f;
    v8f c;
#pragma unroll
    for (int i = 0; i < 8; ++i) c[i] = bias;

    const float* xrow = x + (size_t)(tile + l16) * NFEAT;
    // B source: Wt[k][n] = Wih0[n*124 + k]; mask dead columns to 0
    const int   nidx = ncol ? l16 : 0;
    const float nmsk = ncol ? 1.0f : 0.0f;
    const float* wrow = Wih0 + nidx * NFEAT;

    for (int k0 = 0; k0 < NFEAT; k0 += 4) {
        const int k = k0 + 2 * half;
        // A fragment: lane l16 holds row (tile+l16), K = k, k+1
        v2f a;
        a[0] = xrow[k];
        a[1] = xrow[k + 1];
        // B fragment: lane column n = l16, K split mirrors A
        v2f b;
        b[0] = wrow[k] * nmsk;
        b[1] = wrow[k + 1] * nmsk;
        c = __builtin_amdgcn_wmma_f32_16x16x4_f32(
                /*neg_a=*/false, a, /*neg_b=*/false, b,
                /*c_mod=*/(short)0, c, /*reuse_a=*/false, /*reuse_b=*/false);
    }

    // D layout: VGPR r -> M = tile + r + 8*half, N = l16
    if (ncol) {
#pragma unroll
        for (int r = 0; r < 8; ++r) {
            const int row = tile + r + 8 * half;
            z0[(size_t)row * HID + l16] = c[r];
        }
    }
}

// ---------------------------------------------------------------------------
// Phase 2: sequential scan over T. One lane per batch element (1024 lanes =
// 32 single-wave workgroups). Tiny 3x3 recurrences + fast hardware tanh.
// tanh(x) = 1 - 2 / (exp2(2*log2e*x) + 1); saturates correctly for |x| large.
// ---------------------------------------------------------------------------
__device__ __forceinline__ float fast_tanh(float v) {
    float e = __builtin_amdgcn_exp2f(v * 2.885390081777926815f); // 2*log2(e)
    return 1.0f - 2.0f * __builtin_amdgcn_rcpf(e + 1.0f);
}

__global__ void __launch_bounds__(32)
rnn_phase2_scan(const float* __restrict__ z0,
                const float* __restrict__ Whh0,
                const float* __restrict__ Wih1,
                const float* __restrict__ Whh1,
                const float* __restrict__ bih1,
                const float* __restrict__ bhh1,
                const float* __restrict__ Wfc,
                const float* __restrict__ bfc,
                float* __restrict__ out) {
    const int b = blockIdx.x * blockDim.x + threadIdx.x;
    if (b >= BSZ) return;

    float whh0[9], wih1[9], whh1[9];
#pragma unroll
    for (int i = 0; i < 9; ++i) { whh0[i] = Whh0[i]; wih1[i] = Wih1[i]; whh1[i] = Whh1[i]; }
    float bias1[3];
#pragma unroll
    for (int i = 0; i < 3; ++i) bias1[i] = bih1[i] + bhh1[i];

    float h0[3] = {0.f, 0.f, 0.f};
    float h1[3] = {0.f, 0.f, 0.f};

    const float* zp = z0 + (size_t)b * TLEN * HID;
    for (int t = 0; t < TLEN; ++t) {
        const float z0v = zp[0], z1v = zp[1], z2v = zp[2];
        zp += HID;
        float n0[3];
        n0[0] = fast_tanh(z0v + whh0[0]*h0[0] + whh0[1]*h0[1] + whh0[2]*h0[2]);
        n0[1] = fast_tanh(z1v + whh0[3]*h0[0] + whh0[4]*h0[1] + whh0[5]*h0[2]);
        n0[2] = fast_tanh(z2v + whh0[6]*h0[0] + whh0[7]*h0[1] + whh0[8]*h0[2]);
        float n1[3];
#pragma unroll
        for (int i = 0; i < 3; ++i) {
            n1[i] = fast_tanh(bias1[i]
                              + wih1[3*i]*n0[0] + wih1[3*i+1]*n0[1] + wih1[3*i+2]*n0[2]
                              + whh1[3*i]*h1[0] + whh1[3*i+1]*h1[1] + whh1[3*i+2]*h1[2]);
        }
#pragma unroll
        for (int i = 0; i < 3; ++i) { h0[i] = n0[i]; h1[i] = n1[i]; }
    }

#pragma unroll
    for (int o = 0; o < OUT; ++o) {
        out[(size_t)b * OUT + o] =
            bfc[o] + Wfc[o*3]*h1[0] + Wfc[o*3+1]*h1[1] + Wfc[o*3+2]*h1[2];
    }
}

// ---------------------------------------------------------------------------
extern "C" void kernel_launch(void* const* d_in, const int* in_sizes, int n_in,
                              void* d_out, int out_size, void* d_ws, size_t ws_size,
                              hipStream_t stream) {
    (void)in_sizes; (void)n_in; (void)out_size; (void)ws_size;

    const float* x    = (const float*)d_in[0];
    const float* Wih0 = (const float*)d_in[1];
    const float* Whh0 = (const float*)d_in[2];
    const float* bih0 = (const float*)d_in[3];
    const float* bhh0 = (const float*)d_in[4];
    const float* Wih1 = (const float*)d_in[5];
    const float* Whh1 = (const float*)d_in[6];
    const float* bih1 = (const float*)d_in[7];
    const float* bhh1 = (const float*)d_in[8];
    const float* Wfc  = (const float*)d_in[9];
    const float* bfc  = (const float*)d_in[10];
    float* out = (float*)d_out;
    float* z0  = (float*)d_ws;   // MROWS * 3 floats = 6 MB

    // Phase 1: 524288 rows / (16 rows per wave * 8 waves per block) = 4096 blocks
    rnn_phase1_wmma<<<MROWS / (16 * 8), 256, 0, stream>>>(x, Wih0, bih0, bhh0, z0);

    // Phase 2: 1024 chains, one lane each, 32 single-wave workgroups
    rnn_phase2_scan<<<BSZ / 32, 32, 0, stream>>>(z0, Whh0, Wih1, Whh1,
                                                 bih1, bhh1, Wfc, bfc, out);
}